// SimilarityScaling_55396488183855
// MI455X (gfx1250) — compile-verified
//
#include <hip/hip_runtime.h>
#include <hip/hip_bf16.h>

typedef __attribute__((ext_vector_type(2))) float v2f;
typedef __attribute__((ext_vector_type(4))) float v4f;
typedef __attribute__((ext_vector_type(8))) float v8f;

#define N_PTS 8192
#define DIM   1024
#define K_NN  10
#define TOPK  11      // K_NN + 1 (self included)

// ---------------- GEMM: C[M][N] = A[M][K] * B[N][K]^T (fp32 WMMA) -------------
// Block tile 128(M) x 64(N), 256 threads = 8 waves arranged 4(M) x 2(N);
// each wave computes a 32x32 region as 2x2 tiles of V_WMMA_F32_16X16X4_F32.
// Double-buffered LDS filled by GLOBAL_LOAD_ASYNC_TO_LDS_B128 (ASYNCcnt),
// overlapping the async DMA of chunk i+1 with the WMMA work of chunk i.
#define KC 32
#define LDST 36   // padded row stride (floats): 144B -> 16B aligned, bank spread

__global__ __launch_bounds__(256)
void gemm_abt_f32_wmma(const float* __restrict__ A,
                       const float* __restrict__ B,
                       float* __restrict__ C,
                       int M, int N, int K)
{
    __shared__ float As[2][128 * LDST];
    __shared__ float Bs[2][64  * LDST];

    const int tid   = threadIdx.x;
    const int wave  = tid >> 5;
    const int lane  = tid & 31;
    const int wm    = wave & 3;     // M sub-block (32 rows)
    const int wn    = wave >> 2;    // N sub-block (32 cols)
    const int m16   = lane & 15;
    const int khalf = lane >> 4;    // lanes 0-15: K=0,1 ; lanes 16-31: K=2,3

    const int blockN = blockIdx.x * 64;
    const int blockM = blockIdx.y * 128;

    // staging coordinates: each lane moves one 16B packet per row-slice
    const int rs = tid >> 3;          // 0..31
    const int cs = (tid & 7) * 4;     // float offset within 32-wide K chunk

    const unsigned ldsA0 = (unsigned)(size_t)&As[0][0];
    const unsigned ldsB0 = (unsigned)(size_t)&Bs[0][0];
    const unsigned bufAsz = sizeof(float) * 128 * LDST;
    const unsigned bufBsz = sizeof(float) * 64 * LDST;

    // async-issue one K-chunk into LDS buffer `buf`
    auto issue_chunk = [&](int buf, int k0) {
        #pragma unroll
        for (int i = 0; i < 4; ++i) {                  // A: 128 rows
            int r = rs + i * 32;
            unsigned long long g =
                (unsigned long long)(size_t)(A + (size_t)(blockM + r) * K + k0 + cs);
            unsigned l = ldsA0 + (unsigned)buf * bufAsz + (unsigned)(r * LDST + cs) * 4u;
            asm volatile("global_load_async_to_lds_b128 %0, %1, off"
                         :: "v"(l), "v"(g) : "memory");
        }
        #pragma unroll
        for (int i = 0; i < 2; ++i) {                  // B: 64 rows
            int r = rs + i * 32;
            unsigned long long g =
                (unsigned long long)(size_t)(B + (size_t)(blockN + r) * K + k0 + cs);
            unsigned l = ldsB0 + (unsigned)buf * bufBsz + (unsigned)(r * LDST + cs) * 4u;
            asm volatile("global_load_async_to_lds_b128 %0, %1, off"
                         :: "v"(l), "v"(g) : "memory");
        }
    };

    v8f acc[2][2];
    const v8f vzero = {0.f,0.f,0.f,0.f,0.f,0.f,0.f,0.f};
    acc[0][0] = vzero; acc[0][1] = vzero; acc[1][0] = vzero; acc[1][1] = vzero;

    issue_chunk(0, 0);                 // prologue: fill buffer 0

    const int nchunk = K / KC;
    for (int c = 0; c < nchunk; ++c) {
        // my async loads done -> barrier -> everyone's loads visible in LDS
        asm volatile("s_wait_asynccnt 0x0" ::: "memory");
        __syncthreads();

        if (c + 1 < nchunk)
            issue_chunk((c + 1) & 1, (c + 1) * KC);    // DMA next chunk during compute

        const float* Arow0 = &As[c & 1][(wm * 32 + m16) * LDST + khalf * 2];
        const float* Brow0 = &Bs[c & 1][(wn * 32 + m16) * LDST + khalf * 2];

        #pragma unroll
        for (int k = 0; k < KC; k += 4) {
            v2f a0 = *(const v2f*)(Arow0 + k);
            v2f a1 = *(const v2f*)(Arow0 + k + 16 * LDST);
            v2f b0 = *(const v2f*)(Brow0 + k);
            v2f b1 = *(const v2f*)(Brow0 + k + 16 * LDST);
            acc[0][0] = __builtin_amdgcn_wmma_f32_16x16x4_f32(
                false, a0, false, b0, (short)0, acc[0][0], false, false);
            acc[0][1] = __builtin_amdgcn_wmma_f32_16x16x4_f32(
                false, a0, false, b1, (short)0, acc[0][1], false, false);
            acc[1][0] = __builtin_amdgcn_wmma_f32_16x16x4_f32(
                false, a1, false, b0, (short)0, acc[1][0], false, false);
            acc[1][1] = __builtin_amdgcn_wmma_f32_16x16x4_f32(
                false, a1, false, b1, (short)0, acc[1][1], false, false);
        }
    }

    // ---- epilogue: C/D layout = VGPR r holds M = r (lanes<16) or 8+r (lanes>=16)
    #pragma unroll
    for (int mt = 0; mt < 2; ++mt) {
        #pragma unroll
        for (int nt = 0; nt < 2; ++nt) {
            int col   = blockN + wn * 32 + nt * 16 + m16;
            int rbase = blockM + wm * 32 + mt * 16 + (khalf ? 8 : 0);
            #pragma unroll
            for (int r = 0; r < 8; ++r)
                C[(size_t)(rbase + r) * N + col] = acc[mt][nt][r];
        }
    }
}

// ---------------- row L2-normalize (in place OK) ------------------------------
__global__ __launch_bounds__(256)
void row_normalize(const float* __restrict__ emb, float* __restrict__ x, int D)
{
    __shared__ float red[8];
    __shared__ float invn;
    const int row = blockIdx.x;
    const float* e = emb + (size_t)row * D;
    float s = 0.f;
    for (int i = threadIdx.x; i < D; i += 256) { float v = e[i]; s += v * v; }
    #pragma unroll
    for (int off = 16; off > 0; off >>= 1) s += __shfl_down(s, off, 32);
    if ((threadIdx.x & 31) == 0) red[threadIdx.x >> 5] = s;
    __syncthreads();
    if (threadIdx.x == 0) {
        float t = 0.f;
        #pragma unroll
        for (int i = 0; i < 8; ++i) t += red[i];
        invn = rsqrtf(t);
    }
    __syncthreads();
    const float iv = invn;
    for (int i = threadIdx.x; i < D; i += 256)
        x[(size_t)row * D + i] = e[i] * iv;
}

// ------- per-row top-11 values of S; final[n] = (v2+...+v10)/K ---------------
__global__ __launch_bounds__(256)
void topk_final(const float* __restrict__ S, float* __restrict__ fin, int N)
{
    __shared__ float lists[256 * TOPK];
    const int row = blockIdx.x;
    const float* s = S + (size_t)row * N;

    float best[TOPK];
    #pragma unroll
    for (int i = 0; i < TOPK; ++i) best[i] = -3.0e38f;

    for (int j = threadIdx.x; j < N; j += 256) {
        float v = s[j];
        if (v > best[TOPK - 1]) {
            int p = TOPK - 1;
            while (p > 0 && best[p - 1] < v) { best[p] = best[p - 1]; --p; }
            best[p] = v;
        }
    }
    #pragma unroll
    for (int i = 0; i < TOPK; ++i) lists[threadIdx.x * TOPK + i] = best[i];
    __syncthreads();

    // tree-merge sorted 11-lists
    for (int stride = 128; stride > 0; stride >>= 1) {
        if (threadIdx.x < stride) {
            float* a = &lists[threadIdx.x * TOPK];
            const float* b = &lists[(threadIdx.x + stride) * TOPK];
            float out[TOPK];
            int ia = 0, ib = 0;
            #pragma unroll
            for (int i = 0; i < TOPK; ++i)
                out[i] = (a[ia] >= b[ib]) ? a[ia++] : b[ib++];
            #pragma unroll
            for (int i = 0; i < TOPK; ++i) a[i] = out[i];
        }
        __syncthreads();
    }
    if (threadIdx.x == 0) {
        // ranks 2..10 = sorted indices 1..9 (index 0 is self-similarity ~ 1.0)
        float t = 0.f;
        #pragma unroll
        for (int i = 1; i <= K_NN - 1; ++i) t += lists[i];
        fin[row] = t / (float)K_NN;
    }
}

// ------- dist = 1 - 2*S + final[i] + final[j], in place over d_out -----------
__global__ __launch_bounds__(256)
void dist_fuse(float* __restrict__ S, const float* __restrict__ fin)
{
    const size_t i4   = (size_t)blockIdx.x * 256 + threadIdx.x;
    const size_t base = i4 * 4;
    const int row = (int)(base >> 13);          // /8192
    const int col = (int)(base & 8191);
    v4f v = *(v4f*)(S + base);
    const float fr = fin[row];
    v.x = 1.f - 2.f * v.x + fr + fin[col + 0];
    v.y = 1.f - 2.f * v.y + fr + fin[col + 1];
    v.z = 1.f - 2.f * v.z + fr + fin[col + 2];
    v.w = 1.f - 2.f * v.w + fr + fin[col + 3];
    *(v4f*)(S + base) = v;
}

// -----------------------------------------------------------------------------
extern "C" void kernel_launch(void* const* d_in, const int* in_sizes, int n_in,
                              void* d_out, int out_size, void* d_ws, size_t ws_size,
                              hipStream_t stream)
{
    const float* points = (const float*)d_in[0];   // 8192 x 1024
    const float* W      = (const float*)d_in[1];   // 1024 x 1024
    float* out = (float*)d_out;                    // 8192 x 8192
    float* emb = (float*)d_ws;                     // N*D floats (emb, then x in place)
    float* fin = emb + (size_t)N_PTS * DIM;        // N floats

    // 1) emb = points @ W^T   (M=8192, N=1024, K=1024)
    dim3 g1(DIM / 64, N_PTS / 128);
    gemm_abt_f32_wmma<<<g1, 256, 0, stream>>>(points, W, emb, N_PTS, DIM, DIM);

    // 2) x = normalize_rows(emb), in place
    row_normalize<<<N_PTS, 256, 0, stream>>>(emb, emb, DIM);

    // 3) S = x @ x^T -> d_out  (M=N=8192, K=1024)
    dim3 g2(N_PTS / 64, N_PTS / 128);
    gemm_abt_f32_wmma<<<g2, 256, 0, stream>>>(emb, emb, out, N_PTS, N_PTS, DIM);

    // 4) final[n] = (sum of rank-2..rank-10 values of S row n) / K
    topk_final<<<N_PTS, 256, 0, stream>>>(out, fin, N_PTS);

    // 5) dist = 1 - 2S + final_i + final_j, rewrite d_out in place
    dist_fuse<<<((size_t)N_PTS * N_PTS / 4) / 256, 256, 0, stream>>>(out, fin);
}